// ContrastiveMambaModel_11373073399866
// MI455X (gfx1250) — compile-verified
//
#include <hip/hip_runtime.h>
#include <math.h>
#include <stdint.h>

#define B_     2
#define L_     1024
#define DMODEL 512
#define DINNER 1024
#define DSTATE 16
#define DTRANK 32
#define DCONV  4
#define NLAYER 2
#define EMBDIM 256
#define NEMO   7
#define SCH    64     // selective-scan timestep chunk staged in LDS

typedef __attribute__((ext_vector_type(2))) float v2f;
typedef __attribute__((ext_vector_type(8))) float v8f;
typedef __attribute__((ext_vector_type(4))) unsigned int uint32x4;
typedef __attribute__((ext_vector_type(8))) int int32x8;
typedef __attribute__((ext_vector_type(4))) int int32x4;

__device__ __forceinline__ float sigmoidf_(float x) { return 1.f / (1.f + __expf(-x)); }

// ---------------- embedding gather: x[b,l,dm] = emb[ids[b,l], dm] ----------------
__global__ void embed_kernel(const int* __restrict__ ids, const float* __restrict__ emb,
                             float* __restrict__ x) {
  int i   = blockIdx.x * blockDim.x + threadIdx.x;   // over B*L*DMODEL
  int dm  = i & (DMODEL - 1);
  int row = i >> 9;                                  // DMODEL == 512
  x[i] = emb[(size_t)ids[row] * DMODEL + dm];
}

// ---------------- rmsnorm over rows of length DMODEL (256 thr, 2 elems/thr) ------
__global__ void rmsnorm_kernel(const float* __restrict__ x, const float* __restrict__ w,
                               float* __restrict__ out) {
  __shared__ float red[256];
  int row = blockIdx.x, t = threadIdx.x;
  const float* xr = x + (size_t)row * DMODEL;
  float a = xr[t], b = xr[t + 256];
  red[t] = a * a + b * b;
  __syncthreads();
  for (int o = 128; o > 0; o >>= 1) {
    if (t < o) red[t] += red[t + o];
    __syncthreads();
  }
  float rstd = rsqrtf(red[0] * (1.f / DMODEL) + 1e-5f);
  out[(size_t)row * DMODEL + t]       = a * rstd * w[t];
  out[(size_t)row * DMODEL + t + 256] = b * rstd * w[t + 256];
}

// ---------------- fp32 WMMA GEMM: C = A@B (+Cin) (+bias) (+softplus) -------------
// One wave computes a 16x(16*NT) slab of C via V_WMMA_F32_16X16X4_F32:
// NT independent accumulators share one A fragment per K-step (no WMMA->WMMA RAW).
// Fragment layouts per CDNA5 ISA 7.12.2:
//   A 16x4 : lane l holds A[m=l%16][k = 2*(l/16)+j], j = vgpr 0/1
//   B 4x16 : lane l holds B[k = 2*(l/16)+j][n=l%16]
//   C 16x16: vgpr r, lane l -> C[m = r + 8*(l/16)][n = l%16]
template <int NT>
__global__ void gemm16_wmma(const float* __restrict__ A, int lda,
                            const float* __restrict__ Bm, int ldb,
                            float* __restrict__ C, int ldc,
                            const float* __restrict__ Cin,
                            const float* __restrict__ bias,
                            int K, int act) {
  int lane = threadIdx.x & 31;
  int wave = threadIdx.x >> 5;
  int tm   = blockIdx.x;
  int tn0  = ((blockIdx.y << 2) + wave) * NT;   // first 16-col tile index
  int mr   = lane & 15;
  int g    = lane >> 4;

  const float* Arow = A + (size_t)(tm * 16 + mr) * lda + 2 * g;
  const float* Bp   = Bm + tn0 * 16 + mr + (size_t)(2 * g) * ldb;

  v8f acc[NT];
#pragma unroll
  for (int t = 0; t < NT; ++t) acc[t] = (v8f){};

  for (int k0 = 0; k0 < K; k0 += 4) {
    v2f av = {Arow[k0], Arow[k0 + 1]};
    const float* bk = Bp + (size_t)k0 * ldb;
#pragma unroll
    for (int t = 0; t < NT; ++t) {
      v2f bv = {bk[t * 16], bk[t * 16 + ldb]};
      acc[t] = __builtin_amdgcn_wmma_f32_16x16x4_f32(false, av, false, bv,
                                                     (short)0, acc[t], false, false);
    }
  }

#pragma unroll
  for (int t = 0; t < NT; ++t) {
    int n = (tn0 + t) * 16 + mr;
    float bb = bias ? bias[n] : 0.f;
#pragma unroll
    for (int r = 0; r < 8; ++r) {
      int m = tm * 16 + r + 8 * g;
      float v = acc[t][r] + bb;
      if (Cin) v += Cin[(size_t)m * ldc + n];
      if (act) v = (v > 20.f) ? v : log1pf(__expf(v));   // softplus
      C[(size_t)m * ldc + n] = v;
    }
  }
}

// ---------------- causal depthwise conv1d (k=4) + bias + SiLU --------------------
__global__ void conv_silu_kernel(const float* __restrict__ y1, const float* __restrict__ cw,
                                 const float* __restrict__ cb, float* __restrict__ xi) {
  int i   = blockIdx.x * blockDim.x + threadIdx.x;   // over B*L*DINNER
  int d   = i & (DINNER - 1);
  int row = i >> 10;                                 // DINNER == 1024
  int l   = row & (L_ - 1);
  const float* w = cw + d * DCONV;
  float acc = cb[d];
#pragma unroll
  for (int t = 0; t < DCONV; ++t) {
    int ls = l - (DCONV - 1) + t;
    if (ls >= 0)
      acc += w[t] * y1[(size_t)(row - (DCONV - 1) + t) * (2 * DINNER) + d];
  }
  xi[i] = acc * sigmoidf_(acc);
}

// ---------------- TDM: bulk-stage a [rows x 32] B/C chunk of xdbl into LDS -------
// D# per CDNA5 ISA ch.8: group0 = {count=1, lds_addr, global_addr, type=2},
// group1 = {data_size=4B, tensor_dim0=32, tensor_dim1=rows, tile 32 x rows,
//           tensor_dim0_stride=64}; groups 2/3 zero (2D tensor).
// This toolchain exposes the 6-arg builtin: (g0, g1, g2, g3, g4, cpol).
#if defined(__gfx1250__) && __has_builtin(__builtin_amdgcn_tensor_load_to_lds) && \
    __has_builtin(__builtin_amdgcn_s_wait_tensorcnt)
#define USE_TDM 1
__device__ __forceinline__ void tdm_load_chunk(const float* gsrc, const float* ldst, int rows) {
  uint64_t ga = (uint64_t)(uintptr_t)gsrc;
  uint32_t la = (uint32_t)(uintptr_t)ldst;        // LDS aperture: addr[31:0] = LDS offset
  uint32x4 g0;
  g0[0] = 1u;                                               // count = 1 valid descriptor
  g0[1] = la;                                               // lds_addr
  g0[2] = (uint32_t)ga;                                     // global_addr[31:0]
  g0[3] = ((uint32_t)(ga >> 32) & 0x01FFFFFFu) | 0x80000000u; // global_addr[56:32], type=2
  int32x8 g1;
  g1[0] = 0x00020000;        // data_size = 2 (4 bytes)
  g1[1] = 32 << 16;          // tensor_dim0 = 32
  g1[2] = rows << 16;        // tensor_dim1 = rows
  g1[3] = 32 << 16;          // tile_dim0 = 32
  g1[4] = rows;              // tile_dim1 = rows
  g1[5] = 64;                // tensor_dim0_stride = 64 elements (xdbl row pitch)
  g1[6] = 0;
  g1[7] = 0;
  int32x4 z4 = {0, 0, 0, 0};
  int32x8 z8 = {0, 0, 0, 0, 0, 0, 0, 0};
  __builtin_amdgcn_tensor_load_to_lds(g0, g1, z4, z4, z8, 0);
}
#else
#define USE_TDM 0
#endif

// ---------------- selective scan: thread per (b,d); B/C chunks via TDM + LDS -----
__global__ void scan_kernel(const float* __restrict__ delta, const float* __restrict__ xi,
                            const float* __restrict__ xdbl, const float* __restrict__ Alog,
                            const float* __restrict__ Dp, float* __restrict__ y) {
  __shared__ float sBC[2][SCH * 2 * DSTATE];    // double-buffered [SCH][32]
  int b   = blockIdx.x >> 2;
  int d   = ((blockIdx.x & 3) << 8) + threadIdx.x;
  int tid = threadIdx.x;
  float Ad[DSTATE], h[DSTATE];
#pragma unroll
  for (int n = 0; n < DSTATE; ++n) { Ad[n] = -__expf(Alog[d * DSTATE + n]); h[n] = 0.f; }
  float Dd = Dp[d];
  const int NC = L_ / SCH;
  const float* gbase = xdbl + (size_t)b * L_ * 64 + DTRANK;   // start of B/C columns

#if USE_TDM
  if (tid < 32)                                  // wave 0 drives the DMA engine
    tdm_load_chunk(gbase, sBC[0], SCH);
#endif

  for (int c = 0; c < NC; ++c) {
#if USE_TDM
    if (tid < 32) {
      if (c + 1 < NC) {
        tdm_load_chunk(gbase + (size_t)(c + 1) * SCH * 64, sBC[(c + 1) & 1], SCH);
        __builtin_amdgcn_s_wait_tensorcnt(1);    // chunk c landed; c+1 in flight
      } else {
        __builtin_amdgcn_s_wait_tensorcnt(0);
      }
    }
#else
    for (int u = tid; u < SCH * 2 * DSTATE; u += 256)
      sBC[c & 1][u] = gbase[(size_t)(c * SCH + (u >> 5)) * 64 + (u & 31)];
#endif
    __syncthreads();
    const float* bc = sBC[c & 1];
    for (int j = 0; j < SCH; ++j) {
      int row  = b * L_ + c * SCH + j;
      float dt = delta[(size_t)row * DINNER + d];
      float u  = xi[(size_t)row * DINNER + d];
      const float* bj = bc + j * 2 * DSTATE;
      float yv = 0.f;
#pragma unroll
      for (int n = 0; n < DSTATE; ++n) {
        h[n] = __expf(dt * Ad[n]) * h[n] + dt * bj[n] * u;
        yv  += h[n] * bj[DSTATE + n];
      }
      y[(size_t)row * DINNER + d] = yv + u * Dd;
    }
    __syncthreads();
  }
}

// ---------------- gating: y *= silu(res), res = second half of y1 ----------------
__global__ void gate_kernel(float* __restrict__ y, const float* __restrict__ y1) {
  int i   = blockIdx.x * blockDim.x + threadIdx.x;
  int d   = i & (DINNER - 1);
  int row = i >> 10;
  float r = y1[(size_t)row * (2 * DINNER) + DINNER + d];
  y[i] *= r * sigmoidf_(r);
}

// ---------------- mean pool over L -----------------------------------------------
__global__ void pool_kernel(const float* __restrict__ xn, float* __restrict__ pooled) {
  int i  = blockIdx.x * blockDim.x + threadIdx.x;   // over B*DMODEL
  int b  = i >> 9, dm = i & (DMODEL - 1);
  float s = 0.f;
  for (int l = 0; l < L_; ++l) s += xn[(size_t)(b * L_ + l) * DMODEL + dm];
  pooled[i] = s * (1.f / L_);
}

// ---------------- head: emotion_emb + logits -> d_out ----------------------------
// d_out layout: logits [B,7] first, then emotion_emb [B,256]
__global__ void head_kernel(const float* __restrict__ pooled, const float* __restrict__ pw,
                            const float* __restrict__ pb, const float* __restrict__ cw,
                            const float* __restrict__ cb, float* __restrict__ out) {
  __shared__ float semb[B_ * EMBDIM];
  int t = threadIdx.x;                 // B_*EMBDIM = 512 threads
  int b = t >> 8, e = t & (EMBDIM - 1);
  float acc = pb[e];
  for (int k = 0; k < DMODEL; ++k) acc += pooled[b * DMODEL + k] * pw[k * EMBDIM + e];
  semb[t] = acc;
  out[B_ * NEMO + t] = acc;
  __syncthreads();
  if (t < B_ * NEMO) {
    int bb = t / NEMO, c = t % NEMO;
    float a = cb[c];
    for (int e2 = 0; e2 < EMBDIM; ++e2) a += semb[bb * EMBDIM + e2] * cw[e2 * NEMO + c];
    out[t] = a;
  }
}

extern "C" void kernel_launch(void* const* d_in, const int* in_sizes, int n_in,
                              void* d_out, int out_size, void* d_ws, size_t ws_size,
                              hipStream_t stream) {
  (void)in_sizes; (void)n_in; (void)out_size; (void)ws_size;

  const int*   ids   = (const int*)d_in[0];
  const float* emb   = (const float*)d_in[1];
  const float* normw = (const float*)d_in[2];
  const float* ipw   = (const float*)d_in[3];
  const float* cw    = (const float*)d_in[4];
  const float* cb    = (const float*)d_in[5];
  const float* xpw   = (const float*)d_in[6];
  const float* dtw   = (const float*)d_in[7];
  const float* dtb   = (const float*)d_in[8];
  const float* alog  = (const float*)d_in[9];
  const float* Dp    = (const float*)d_in[10];
  const float* opw   = (const float*)d_in[11];
  const float* nfw   = (const float*)d_in[12];
  const float* pw    = (const float*)d_in[13];
  const float* pb    = (const float*)d_in[14];
  const float* clw   = (const float*)d_in[15];
  const float* clb   = (const float*)d_in[16];

  const size_t NTOK = (size_t)B_ * L_;              // 2048 rows
  float* ws    = (float*)d_ws;
  float* x     = ws;                                // NTOK*DMODEL
  float* xn    = x     + NTOK * DMODEL;             // NTOK*DMODEL
  float* y1    = xn    + NTOK * DMODEL;             // NTOK*2*DINNER
  float* xiact = y1    + NTOK * 2 * DINNER;         // NTOK*DINNER
  float* xdbl  = xiact + NTOK * DINNER;             // NTOK*64
  float* dlt   = xdbl  + NTOK * 64;                 // NTOK*DINNER
  float* yscan = dlt   + NTOK * DINNER;             // NTOK*DINNER
  float* pooled= yscan + NTOK * DINNER;             // B_*DMODEL

  embed_kernel<<<(NTOK * DMODEL) / 256, 256, 0, stream>>>(ids, emb, x);

  for (int layer = 0; layer < NLAYER; ++layer) {
    const float* nw_l  = normw + layer * DMODEL;
    const float* ipw_l = ipw + (size_t)layer * DMODEL * 2 * DINNER;
    const float* cw_l  = cw  + (size_t)layer * DINNER * DCONV;
    const float* cb_l  = cb  + layer * DINNER;
    const float* xpw_l = xpw + (size_t)layer * DINNER * (DTRANK + 2 * DSTATE);
    const float* dtw_l = dtw + (size_t)layer * DTRANK * DINNER;
    const float* dtb_l = dtb + layer * DINNER;
    const float* al_l  = alog+ (size_t)layer * DINNER * DSTATE;
    const float* D_l   = Dp  + layer * DINNER;
    const float* opw_l = opw + (size_t)layer * DINNER * DMODEL;

    // x -> rmsnorm -> xn
    rmsnorm_kernel<<<NTOK, 256, 0, stream>>>(x, nw_l, xn);
    // y1[2048,2048] = xn[2048,512] @ in_proj_w[512,2048]   (NT=4 -> 256 cols/block)
    gemm16_wmma<4><<<dim3(NTOK / 16, (2 * DINNER) / 256), 128, 0, stream>>>(
        xn, DMODEL, ipw_l, 2 * DINNER, y1, 2 * DINNER, nullptr, nullptr, DMODEL, 0);
    // causal depthwise conv + SiLU on xi half -> xiact
    conv_silu_kernel<<<(NTOK * DINNER) / 256, 256, 0, stream>>>(y1, cw_l, cb_l, xiact);
    // x_dbl[2048,64] = xiact @ x_proj_w[1024,64]           (NT=1 -> 64 cols/block)
    gemm16_wmma<1><<<dim3(NTOK / 16, 1), 128, 0, stream>>>(
        xiact, DINNER, xpw_l, 64, xdbl, 64, nullptr, nullptr, DINNER, 0);
    // delta[2048,1024] = softplus(x_dbl[:, :32] @ dt_proj_w[32,1024] + dt_proj_b)
    gemm16_wmma<4><<<dim3(NTOK / 16, DINNER / 256), 128, 0, stream>>>(
        xdbl, 64, dtw_l, DINNER, dlt, DINNER, nullptr, dtb_l, DTRANK, 1);
    // selective scan -> yscan (TDM-staged B/C chunks)
    scan_kernel<<<B_ * 4, 256, 0, stream>>>(dlt, xiact, xdbl, al_l, D_l, yscan);
    // yscan *= silu(res)
    gate_kernel<<<(NTOK * DINNER) / 256, 256, 0, stream>>>(yscan, y1);
    // x += yscan @ out_proj_w[1024,512]   (residual fused via Cin)
    gemm16_wmma<4><<<dim3(NTOK / 16, DMODEL / 256), 128, 0, stream>>>(
        yscan, DINNER, opw_l, DMODEL, x, DMODEL, x, nullptr, DINNER, 0);
  }

  rmsnorm_kernel<<<NTOK, 256, 0, stream>>>(x, nfw, xn);
  pool_kernel<<<(B_ * DMODEL) / 256, 256, 0, stream>>>(xn, pooled);
  head_kernel<<<1, B_ * EMBDIM, 0, stream>>>(pooled, pw, pb, clw, clb, (float*)d_out);
}